// FCGF_point_att3_89575837925659
// MI455X (gfx1250) — compile-verified
//
#include <hip/hip_runtime.h>
#include <hip/hip_bf16.h>

typedef __attribute__((ext_vector_type(2))) float v2f;
typedef __attribute__((ext_vector_type(8))) float v8f;

#define NPTS   800000
#define PTS    50000
#define NSEG   16

// workspace layout (floats)
#define OFF_S     0        // 32x32 = 1024
#define OFF_SEGX  1024     // 16x32 = 512
#define OFF_SEGXA 1536     // 16x32 = 512
#define OFF_SUMA  2048
#define OFF_SUMAA 2049
#define OFF_W1E   2056     // 16x32 = 512
#define OFF_B1E   2568     // 16
#define WS_FLOATS 2592

__device__ __forceinline__ v8f wmma4(v2f a, v2f b, v8f c) {
  // D = A(16x4 f32) x B(4x16 f32) + C(16x16 f32)
  return __builtin_amdgcn_wmma_f32_16x16x4_f32(
      /*neg_a=*/false, a, /*neg_b=*/false, b,
      /*c_mod=*/(short)0, c, /*reuse_a=*/false, /*reuse_b=*/false);
}

// ---------------- Pass 1: S = X^T X (32x32) and per-segment sum(x) ----------------
// grid = 16 segs * 10 blocks, 512 threads (16 waves). 5000 pts/block, quads of 4.
__global__ void pass1_xtx(const float* __restrict__ x, float* __restrict__ ws) {
  __shared__ float Sp[1024];
  __shared__ float segx_s[32];
  const int t = threadIdx.x;
  Sp[t] = 0.f; Sp[t + 512] = 0.f;
  if (t < 32) segx_s[t] = 0.f;
  __syncthreads();

  const int seg  = blockIdx.x / 10;
  const int blk  = blockIdx.x % 10;
  const long base = (long)seg * PTS + (long)blk * 5000;
  const int wave = t >> 5, lane = t & 31;
  const int c = lane & 15, half = lane >> 4;

  v8f acc00 = {}, acc01 = {}, acc10 = {}, acc11 = {};
  float sx0 = 0.f, sx1 = 0.f;

  for (int q = wave; q < 1250; q += 16) {            // 4 points per step
    const float* row = x + (base + (long)q * 4 + 2 * half) * 32 + c;
    float v00 = row[0];    // x[k0+2h  ][c]
    float v01 = row[32];   // x[k0+2h+1][c]
    float v10 = row[16];   // x[k0+2h  ][c+16]
    float v11 = row[48];   // x[k0+2h+1][c+16]
    v2f A0; A0.x = v00; A0.y = v01;
    v2f A1; A1.x = v10; A1.y = v11;
    // same registers serve as A-operand (rows) and B-operand (cols)
    acc00 = wmma4(A0, A0, acc00);
    acc01 = wmma4(A0, A1, acc01);
    acc10 = wmma4(A1, A0, acc10);
    acc11 = wmma4(A1, A1, acc11);
    sx0 += v00 + v01;
    sx1 += v10 + v11;
  }

  // C layout: reg r, lane -> M = r + 8*half, Ncol = c (within tile)
#pragma unroll
  for (int r = 0; r < 8; ++r) {
    int M0 = r + 8 * half;
    atomicAdd(&Sp[M0 * 32 + c],            acc00[r]);
    atomicAdd(&Sp[M0 * 32 + 16 + c],       acc01[r]);
    atomicAdd(&Sp[(M0 + 16) * 32 + c],     acc10[r]);
    atomicAdd(&Sp[(M0 + 16) * 32 + 16 + c], acc11[r]);
  }
  sx0 += __shfl_xor(sx0, 16, 32);
  sx1 += __shfl_xor(sx1, 16, 32);
  if (lane < 16) {
    atomicAdd(&segx_s[c], sx0);
    atomicAdd(&segx_s[c + 16], sx1);
  }
  __syncthreads();
  for (int idx = t; idx < 1024; idx += 512)
    unsafeAtomicAdd(&ws[OFF_S + idx], Sp[idx]);
  if (t < 32)
    unsafeAtomicAdd(&ws[OFF_SEGX + seg * 32 + t], segx_s[t]);
}

// ---------------- Fold BN1 into affine W1eff/b1eff from S, sum(x) ----------------
__global__ void fold_bn1(const float* __restrict__ W1, const float* __restrict__ b1,
                         const float* __restrict__ g1, const float* __restrict__ be1,
                         float* __restrict__ ws) {
  __shared__ float sxt[32];
  const int t = threadIdx.x;
  if (t < 32) {
    float s = 0.f;
    for (int b = 0; b < NSEG; ++b) s += ws[OFF_SEGX + b * 32 + t];
    sxt[t] = s;
  }
  __syncthreads();
  if (t < 16) {
    const float invN = 1.0f / (float)NPTS;
    float dot = 0.f;
    for (int k = 0; k < 32; ++k) dot += W1[t * 32 + k] * sxt[k];
    float quad = 0.f;
    for (int i = 0; i < 32; ++i) {
      float wi = W1[t * 32 + i];
      for (int j = 0; j < 32; ++j) quad += wi * W1[t * 32 + j] * ws[OFF_S + i * 32 + j];
    }
    float bb   = b1[t];
    float mean = dot * invN + bb;
    float e2   = quad * invN + 2.f * bb * dot * invN + bb * bb;
    float var  = e2 - mean * mean;
    float inv  = rsqrtf(var + 1e-5f) * g1[t];
    for (int k = 0; k < 32; ++k) ws[OFF_W1E + t * 32 + k] = W1[t * 32 + k] * inv;
    ws[OFF_B1E + t] = (bb - mean) * inv + be1[t];
  }
}

// ---------------- Pass 2: a_raw per point, accumulate segsum(x*a), sum a, sum a^2 --
// grid = 16 segs * 25 blocks, 256 threads (8 waves). 2000 pts/block, tiles of 16.
__global__ void pass2_att(const float* __restrict__ x, const float* __restrict__ W2,
                          const float* __restrict__ b2, float* __restrict__ ws) {
  const int t = threadIdx.x;
  const int seg = blockIdx.x / 25;
  const int blk = blockIdx.x % 25;
  const long base = (long)seg * PTS + (long)blk * 2000;
  const int wave = t >> 5, lane = t & 31;
  const int c = lane & 15, half = lane >> 4;

  // B operand: W1eff^T, K=32 in 8 steps of 4.  breg[k] = W1eff[c][4k+2h .. +1]
  v2f breg[8];
#pragma unroll
  for (int k = 0; k < 8; ++k) {
    const float* wr = ws + OFF_W1E + c * 32 + k * 4 + 2 * half;
    v2f bk; bk.x = wr[0]; bk.y = wr[1];
    breg[k] = bk;
  }
  const float b1e = ws[OFF_B1E + c];
  const float w2c = W2[c];
  const float b2s = b2[0];

  float xaacc[8][2] = {};
  float suma = 0.f, sumaa = 0.f;

  for (int tile = wave; tile < 125; tile += 8) {
    const long p0 = base + (long)tile * 16;
    const float* row = x + (p0 + c) * 32 + 2 * half;
    v2f areg[8];
#pragma unroll
    for (int k = 0; k < 8; ++k) areg[k] = *(const v2f*)(row + k * 4);

    v8f acc = {};
#pragma unroll
    for (int k = 0; k < 8; ++k) acc = wmma4(areg[k], breg[k], acc);

    // acc[r]: point p0 + r + 8*half, channel c.  ReLU(BN1) * W2
    float pa[8];
#pragma unroll
    for (int r = 0; r < 8; ++r) {
      float h = fmaxf(acc[r] + b1e, 0.f);
      pa[r] = h * w2c;
    }
    // reduce over 16 channels within each half-wave
#pragma unroll
    for (int m = 1; m < 16; m <<= 1)
#pragma unroll
      for (int r = 0; r < 8; ++r) pa[r] += __shfl_xor(pa[r], m, 32);
#pragma unroll
    for (int r = 0; r < 8; ++r) pa[r] += b2s;   // a_raw for point p0 + 8*half + r

    // lane wants a_raw of point j = c: select pa[c&7], fetch from half (c>>3)
    float s0 = (c & 1) ? pa[1] : pa[0];
    float s1 = (c & 1) ? pa[3] : pa[2];
    float s2 = (c & 1) ? pa[5] : pa[4];
    float s3 = (c & 1) ? pa[7] : pa[6];
    float u0 = (c & 2) ? s1 : s0;
    float u1 = (c & 2) ? s3 : s2;
    float asel = (c & 4) ? u1 : u0;
    int srcLane = ((c >> 3) << 4) | (c & 7);
    float aj = __shfl(asel, srcLane, 32);       // a_raw[p0 + c]

    // x*a reuses A registers: areg[k] = x[p0+c][4k+2h .. +1]
#pragma unroll
    for (int k = 0; k < 8; ++k) {
      xaacc[k][0] += areg[k].x * aj;
      xaacc[k][1] += areg[k].y * aj;
    }
    if (half == 0) { suma += aj; sumaa += aj * aj; }
  }

  // columns 4k+2h+s are identical across the 16 lanes of a half -> butterfly
#pragma unroll
  for (int m = 1; m < 16; m <<= 1)
#pragma unroll
    for (int k = 0; k < 8; ++k) {
      xaacc[k][0] += __shfl_xor(xaacc[k][0], m, 32);
      xaacc[k][1] += __shfl_xor(xaacc[k][1], m, 32);
    }
  if (c == 0) {
#pragma unroll
    for (int k = 0; k < 8; ++k) {
      unsafeAtomicAdd(&ws[OFF_SEGXA + seg * 32 + k * 4 + 2 * half + 0], xaacc[k][0]);
      unsafeAtomicAdd(&ws[OFF_SEGXA + seg * 32 + k * 4 + 2 * half + 1], xaacc[k][1]);
    }
  }
#pragma unroll
  for (int m = 1; m < 16; m <<= 1) {
    suma  += __shfl_xor(suma, m, 32);
    sumaa += __shfl_xor(sumaa, m, 32);
  }
  if (lane == 0) {
    unsafeAtomicAdd(&ws[OFF_SUMA], suma);
    unsafeAtomicAdd(&ws[OFF_SUMAA], sumaa);
  }
}

// ---------------- Finalize: fold BN2, divide by counts, L2 normalize ----------------
__global__ void finalize(const float* __restrict__ ws, const int* __restrict__ length,
                         const float* __restrict__ g2, const float* __restrict__ be2,
                         float* __restrict__ out) {
  const int t = threadIdx.x;
  const int b = t >> 5, ch = t & 31;
  const float N = (float)NPTS;
  float m2 = ws[OFF_SUMA] / N;
  float v2 = ws[OFF_SUMAA] / N - m2 * m2;
  float inv2  = rsqrtf(v2 + 1e-5f) * g2[0];
  float alpha = inv2;
  float beta  = be2[0] - m2 * inv2;
  float val = (alpha * ws[OFF_SEGXA + b * 32 + ch] + beta * ws[OFF_SEGX + b * 32 + ch])
              / (float)length[b];
  float ss = val * val;
#pragma unroll
  for (int m = 1; m < 32; m <<= 1) ss += __shfl_xor(ss, m, 32);
  float norm = sqrtf(ss);
  out[b * 32 + ch] = val / fmaxf(norm, 1e-12f);
}

extern "C" void kernel_launch(void* const* d_in, const int* in_sizes, int n_in,
                              void* d_out, int out_size, void* d_ws, size_t ws_size,
                              hipStream_t stream) {
  (void)in_sizes; (void)n_in; (void)out_size; (void)ws_size;
  const float* x   = (const float*)d_in[0];
  const float* W1  = (const float*)d_in[1];
  const float* b1  = (const float*)d_in[2];
  const float* g1  = (const float*)d_in[3];
  const float* be1 = (const float*)d_in[4];
  const float* W2  = (const float*)d_in[5];
  const float* b2  = (const float*)d_in[6];
  const float* g2  = (const float*)d_in[7];
  const float* be2 = (const float*)d_in[8];
  const int* length = (const int*)d_in[10];
  float* ws  = (float*)d_ws;
  float* out = (float*)d_out;

  hipMemsetAsync(d_ws, 0, WS_FLOATS * sizeof(float), stream);
  pass1_xtx<<<NSEG * 10, 512, 0, stream>>>(x, ws);
  fold_bn1<<<1, 64, 0, stream>>>(W1, b1, g1, be1, ws);
  pass2_att<<<NSEG * 25, 256, 0, stream>>>(x, W2, b2, ws);
  finalize<<<1, 512, 0, stream>>>(ws, length, g2, be2, out);
}